// PlaneStochastic_42502996361361
// MI455X (gfx1250) — compile-verified
//
#include <hip/hip_runtime.h>

// Sinkhorn (multi-axis masked normalization), linear-domain reformulation.
// B=128, N1=N2=512, fp32, MAX_ITER=10, TAU=1.
//
// p0 = mask ? exp(t) : 0
// repeat 10x: p /= rowsum(p) ; p /= colsum(p)
// output = p  (zeros outside mask preserved exactly by construction)

typedef float v2f __attribute__((ext_vector_type(2)));
typedef float v8f __attribute__((ext_vector_type(8)));

#define NB   128
#define NN   512
#define TAU  1.0f

// ---------------------------------------------------------------------------
// Kernel 1: p = mask ? exp(t / TAU) : 0      (one float4 per thread)
// grid: (NN*NN/1024, NB), block: 256
// ---------------------------------------------------------------------------
__global__ void sinkhorn_init(const float* __restrict__ t,
                              const int* __restrict__ ns,
                              float* __restrict__ p) {
    const int b  = blockIdx.y;
    const int n1 = ns[b];        // ns_list[0][b] : valid rows
    const int n2 = ns[NB + b];   // ns_list[1][b] : valid cols

    const int idx = (blockIdx.x * blockDim.x + threadIdx.x) << 2; // elem in batch
    const int r = idx >> 9;      // / 512
    const int c = idx & (NN - 1);

    const size_t base = (size_t)b * NN * NN + idx;
    const float4 v = *(const float4*)(t + base);

    float4 o;
    const bool rok = (r < n1);
    o.x = (rok && (c + 0) < n2) ? __expf(v.x * (1.0f / TAU)) : 0.0f;
    o.y = (rok && (c + 1) < n2) ? __expf(v.y * (1.0f / TAU)) : 0.0f;
    o.z = (rok && (c + 2) < n2) ? __expf(v.z * (1.0f / TAU)) : 0.0f;
    o.w = (rok && (c + 3) < n2) ? __expf(v.w * (1.0f / TAU)) : 0.0f;

    *(float4*)(p + base) = o;
}

// ---------------------------------------------------------------------------
// Kernel 2: row normalize (sum over axis 2). One wave per 512-float row.
// grid: (NN/8, NB), block: 256 (8 waves)
// Cross-lane reduction via v_wmma_f32_16x16x4_f32 on the matrix pipe:
//   A.vgpr0 = per-lane partial, A.vgpr1 = 0, B = ones
//   => D[m,n] = s_m + s_{m+16}; sum of the 8 C/D VGPRs gives a half-group
//   sum in every lane; one xor-16 shuffle completes the 32-lane total.
// ---------------------------------------------------------------------------
__global__ void sinkhorn_rownorm(float* __restrict__ p,
                                 const int* __restrict__ ns) {
    const int b    = blockIdx.y;
    const int wave = threadIdx.x >> 5;
    const int lane = threadIdx.x & 31;
    const int row  = blockIdx.x * 8 + wave;

    float* rowp = p + ((size_t)b * NN + row) * NN;

    const int c0 = lane << 2;   // 4 coalesced float4 chunks, 128 apart
    float4 v0 = *(const float4*)(rowp + c0 + 0);
    float4 v1 = *(const float4*)(rowp + c0 + 128);
    float4 v2 = *(const float4*)(rowp + c0 + 256);
    float4 v3 = *(const float4*)(rowp + c0 + 384);

    float s = (((v0.x + v0.y) + (v0.z + v0.w)) + ((v1.x + v1.y) + (v1.z + v1.w)))
            + (((v2.x + v2.y) + (v2.z + v2.w)) + ((v3.x + v3.y) + (v3.z + v3.w)));

#if __has_builtin(__builtin_amdgcn_wmma_f32_16x16x4_f32)
    v2f a;  a.x = s;     a.y = 0.0f;
    v2f bo; bo.x = 1.0f; bo.y = 1.0f;
    v8f acc = {};
    // (neg_a, A, neg_b, B, c_mod, C, reuse_a, reuse_b)
    acc = __builtin_amdgcn_wmma_f32_16x16x4_f32(false, a, false, bo,
                                                (short)0, acc, false, false);
    float half = ((acc[0] + acc[1]) + (acc[2] + acc[3]))
               + ((acc[4] + acc[5]) + (acc[6] + acc[7]));
    float total = half + __shfl_xor(half, 16, 32);
#else
    float total = s;
    #pragma unroll
    for (int off = 16; off > 0; off >>= 1) total += __shfl_xor(total, off, 32);
#endif

    // rows beyond n1 are all-zero => total==0 => scale 0 keeps them zero.
    const float scale = (total > 0.0f) ? (1.0f / total) : 0.0f;

    v0.x *= scale; v0.y *= scale; v0.z *= scale; v0.w *= scale;
    v1.x *= scale; v1.y *= scale; v1.z *= scale; v1.w *= scale;
    v2.x *= scale; v2.y *= scale; v2.z *= scale; v2.w *= scale;
    v3.x *= scale; v3.y *= scale; v3.z *= scale; v3.w *= scale;

    *(float4*)(rowp + c0 + 0)   = v0;
    *(float4*)(rowp + c0 + 128) = v1;
    *(float4*)(rowp + c0 + 256) = v2;
    *(float4*)(rowp + c0 + 384) = v3;
}

// ---------------------------------------------------------------------------
// Kernel 3: column normalize (sum over axis 1). One block (512 threads,
// 16 waves) per batch; thread <-> column, fully coalesced row streaming.
// Only the n1 valid rows carry nonzeros, so restrict traffic to them.
// ---------------------------------------------------------------------------
__global__ void sinkhorn_colnorm(float* __restrict__ p,
                                 const int* __restrict__ ns) {
    const int b  = blockIdx.x;
    const int c  = threadIdx.x;
    const int n1 = ns[b];

    float* bp = p + (size_t)b * NN * NN;

    float a0 = 0.0f, a1 = 0.0f, a2 = 0.0f, a3 = 0.0f;
    int r = 0;
    for (; r + 3 < n1; r += 4) {
        __builtin_prefetch(bp + (size_t)(r + 16) * NN + c, 0, 0);
        a0 += bp[(size_t)(r + 0) * NN + c];
        a1 += bp[(size_t)(r + 1) * NN + c];
        a2 += bp[(size_t)(r + 2) * NN + c];
        a3 += bp[(size_t)(r + 3) * NN + c];
    }
    for (; r < n1; ++r) a0 += bp[(size_t)r * NN + c];

    const float total = (a0 + a1) + (a2 + a3);
    const float scale = (total > 0.0f) ? (1.0f / total) : 0.0f;

    for (r = 0; r < n1; ++r) {
        bp[(size_t)r * NN + c] *= scale;
    }
    // rows >= n1 stay exactly 0 from init; never touched.
}

// ---------------------------------------------------------------------------
// Host-side orchestration: 1 init + 10 x (row pass, col pass), all on stream.
// Works in place in d_out (exactly the tensor size); no workspace needed.
// ---------------------------------------------------------------------------
extern "C" void kernel_launch(void* const* d_in, const int* in_sizes, int n_in,
                              void* d_out, int out_size, void* d_ws, size_t ws_size,
                              hipStream_t stream) {
    const float* t  = (const float*)d_in[0];
    const int*   ns = (const int*)d_in[1];   // shape (2, 128) int32
    float*       p  = (float*)d_out;

    (void)in_sizes; (void)n_in; (void)out_size; (void)d_ws; (void)ws_size;

    sinkhorn_init<<<dim3((NN * NN) / 1024, NB), 256, 0, stream>>>(t, ns, p);

    for (int it = 0; it < 10; ++it) {
        sinkhorn_rownorm<<<dim3(NN / 8, NB), 256, 0, stream>>>(p, ns);
        sinkhorn_colnorm<<<dim3(NB), NN, 0, stream>>>(p, ns);
    }
}